// GCNConvBase_38019050504324
// MI455X (gfx1250) — compile-verified
//
#include <hip/hip_runtime.h>

typedef float v2f __attribute__((ext_vector_type(2)));
typedef float v8f __attribute__((ext_vector_type(8)));

#define DIM 128   // DIM_IN == DIM_OUT == 128

// ---------------------------------------------------------------------------
// Kernel 1: out[n][d] = b[d]   (broadcast bias, vectorized float4)
// ---------------------------------------------------------------------------
__global__ void gcn_bias_init_kernel(const float* __restrict__ b,
                                     float* __restrict__ out, int n4) {
    int i = blockIdx.x * blockDim.x + threadIdx.x;
    if (i < n4) {
        const float4* b4 = (const float4*)b;      // 32 float4 per 128-dim row
        ((float4*)out)[i] = b4[i & 31];
    }
}

// ---------------------------------------------------------------------------
// Kernel 2: h = x @ W  via v_wmma_f32_16x16x4_f32 (full fp32 WMMA).
// Block = 256 threads = 8 waves. Block -> one 16-row stripe of x;
// wave w -> 16-col tile w. 32 K-steps of 16x16x4 per tile.
//
// fp32 WMMA operand layout (ISA 7.12.2, wave32):
//   A 16x4 : lane {m=lane&15, kh=lane>>4}; VGPR v holds A[m][4s + 2*kh + v]
//   B 4x16 : lane {n=lane&15, kh=lane>>4}; VGPR v holds B[4s + 2*kh + v][n]
//   C 16x16: VGPR r, lane l -> C[r + 8*(l>>4)][l&15]
// ---------------------------------------------------------------------------
__global__ void gcn_gemm_wmma_kernel(const float* __restrict__ x,
                                     const float* __restrict__ W,
                                     float* __restrict__ h, int n_rows) {
    const int lane = threadIdx.x & 31;
    const int wave = threadIdx.x >> 5;      // 0..7 -> column tile
    const int m    = lane & 15;
    const int half = lane >> 4;

    const int row0 = blockIdx.x * 16;
    const int col0 = wave * 16;
    if (row0 >= n_rows) return;

    const float* xrow = x + (size_t)(row0 + m) * DIM;   // A row for this lane
    const float* wcol = W + col0 + m;                   // B column for this lane

    v8f acc = {};
#pragma unroll 8
    for (int s = 0; s < 32; ++s) {
        const int k = s * 4 + half * 2;
        v2f a;
        a.x = xrow[k];
        a.y = xrow[k + 1];
        v2f bf;
        bf.x = wcol[(size_t)k * DIM];
        bf.y = wcol[(size_t)(k + 1) * DIM];
        // (neg_a, A, neg_b, B, c_mod, C, reuse_a, reuse_b)
        acc = __builtin_amdgcn_wmma_f32_16x16x4_f32(
            false, a, false, bf, (short)0, acc, false, false);
    }

    float* hout = h + (size_t)row0 * DIM + col0 + m;
#pragma unroll
    for (int r = 0; r < 8; ++r) {
        hout[(size_t)(r + 8 * half) * DIM] = acc[r];
    }
}

// ---------------------------------------------------------------------------
// Kernel 3: scatter-add over edges. One wave32 per edge: lane holds float4,
// 32 lanes x 16B == one full 512B feature row (perfectly coalesced gather
// from L2-resident h), then 4 global_atomic_add_f32 per lane into out[dst].
// ---------------------------------------------------------------------------
__global__ void gcn_edge_scatter_kernel(const float* __restrict__ h,
                                        const int* __restrict__ src,
                                        const int* __restrict__ dst,
                                        float* __restrict__ out,
                                        int n_edges, int n_nodes) {
    const int gwave  = (int)((blockIdx.x * blockDim.x + threadIdx.x) >> 5);
    const int lane   = threadIdx.x & 31;
    const int nwaves = (int)((gridDim.x * blockDim.x) >> 5);

    for (int e = gwave; e < n_edges; e += nwaves) {
        const int s = src[e];
        const int d = dst[e];
        if ((unsigned)s >= (unsigned)n_nodes || (unsigned)d >= (unsigned)n_nodes)
            continue;
        const float4 v = ((const float4*)(h + (size_t)s * DIM))[lane];
        float* o = out + (size_t)d * DIM + lane * 4;
        atomicAdd(o + 0, v.x);
        atomicAdd(o + 1, v.y);
        atomicAdd(o + 2, v.z);
        atomicAdd(o + 3, v.w);
    }
}

// ---------------------------------------------------------------------------
// Inputs (setup_inputs order):
//   0: x   [100000*128] f32      1: edge_index [2*1600000] int
//   2: edge_attr (ignored)       3: return_attention_weights (ignored)
//   4: W   [128*128] f32         5: b [128] f32
// d_out: [100000*128] f32. d_ws: >= 51.2 MB for h.
// ---------------------------------------------------------------------------
extern "C" void kernel_launch(void* const* d_in, const int* in_sizes, int n_in,
                              void* d_out, int out_size, void* d_ws, size_t ws_size,
                              hipStream_t stream) {
    const float* x  = (const float*)d_in[0];
    const int*   ei = (const int*)d_in[1];
    const float* W  = (const float*)d_in[4];
    const float* b  = (const float*)d_in[5];

    const int n_nodes = in_sizes[0] / DIM;
    const int n_edges = in_sizes[1] / 2;
    const int* src = ei;
    const int* dst = ei + n_edges;

    float* out = (float*)d_out;
    float* h   = (float*)d_ws;                 // n_nodes * DIM floats

    // 1) out = bias broadcast
    const int n4 = (n_nodes * DIM) / 4;
    gcn_bias_init_kernel<<<(n4 + 255) / 256, 256, 0, stream>>>(b, out, n4);

    // 2) h = x @ W   (one block per 16-row stripe; 8 waves = 8 col tiles)
    const int row_tiles = n_nodes / 16;        // 100000 / 16 = 6250 exact
    gcn_gemm_wmma_kernel<<<row_tiles, 256, 0, stream>>>(x, W, h, n_nodes);

    // 3) out[dst] += h[src]   (one wave per edge, 8 waves per block)
    const int wave_blocks = (n_edges + 7) / 8;
    gcn_edge_scatter_kernel<<<wave_blocks, 256, 0, stream>>>(
        h, src, dst, out, n_edges, n_nodes);
}